// QuantumLayer_52037823758458
// MI455X (gfx1250) — compile-verified
//
#include <hip/hip_runtime.h>

// Elementwise out[i] = cos(x[i]) over 33.5M fp32 elements.
// Memory-bound: 256 MB total traffic -> ~11us floor at 23.3 TB/s.
// Straight-line kernel: one float4 (b128) NT load -> 4x v_cos_f32 -> one
// b128 NT store per thread. Grid exactly covers the data; no grid-stride
// bookkeeping in the hot path.

typedef float v4f __attribute__((ext_vector_type(4)));

#define INV_2PI 0.15915494309189535f  // v_cos_f32 input is in revolutions

__global__ __launch_bounds__(256) void cos_elemwise_kernel(
    const float* __restrict__ x, float* __restrict__ out, int n4, int n) {
  int i = blockIdx.x * blockDim.x + threadIdx.x;

  if (i < n4) {
    const v4f* __restrict__ xin = (const v4f*)x;
    v4f* __restrict__ o = (v4f*)out;
    v4f v = __builtin_nontemporal_load(xin + i);
    v4f r;
    r.x = __builtin_amdgcn_cosf(v.x * INV_2PI);
    r.y = __builtin_amdgcn_cosf(v.y * INV_2PI);
    r.z = __builtin_amdgcn_cosf(v.z * INV_2PI);
    r.w = __builtin_amdgcn_cosf(v.w * INV_2PI);
    __builtin_nontemporal_store(r, o + i);
  }

  // Scalar tail (only if n % 4 != 0; not taken for this problem size).
  int t = n4 * 4 + i;
  if (t < n) {
    float v = __builtin_nontemporal_load(x + t);
    __builtin_nontemporal_store(__builtin_amdgcn_cosf(v * INV_2PI), out + t);
  }
}

extern "C" void kernel_launch(void* const* d_in, const int* in_sizes, int n_in,
                              void* d_out, int out_size, void* d_ws, size_t ws_size,
                              hipStream_t stream) {
  const float* x = (const float*)d_in[0];
  float* out = (float*)d_out;

  int n = out_size;        // 33,554,432 elements
  int n4 = n / 4;          // 8,388,608 float4s

  const int threads = 256; // 8 wave32s per block
  int blocks = (n4 + threads - 1) / threads;  // 32,768 blocks, exact cover
  if (blocks < 1) blocks = 1;

  cos_elemwise_kernel<<<blocks, threads, 0, stream>>>(x, out, n4, n);
}